// LinearizationLayer_62775241999044
// MI455X (gfx1250) — compile-verified
//
#include <hip/hip_runtime.h>

typedef __attribute__((ext_vector_type(2))) float v2f;
typedef __attribute__((ext_vector_type(8))) float v8f;

#define MPAD 5008   // M=5000 padded to multiple of 16; LDS = (2+1)*5008*4 = 60,096 B
#define QPB  128    // queries per block: 8 waves * 16 rows

__global__ __launch_bounds__(256) void LinearizationLayer_nn_wmma(
    const float* __restrict__ ed,   // euclidean_data [N,2]
    const float* __restrict__ mp,   // maze_points    [M,2]
    const float* __restrict__ ts,   // ts_proj        [M]
    float* __restrict__ out,        // [N*2] projected_pos ++ [N] linear_pos
    int N, int M)
{
    __shared__ __align__(16) float s_pk[2 * MPAD];  // interleaved (x,y)
    __shared__ float s_mn[MPAD];                    // |m|^2 (3e38 pad never wins)

    const int tid = threadIdx.x;
    for (int j = tid; j < MPAD; j += 256) {
        if (j < M) {
            float x = mp[2 * j + 0];
            float y = mp[2 * j + 1];
            s_pk[2 * j + 0] = x;
            s_pk[2 * j + 1] = y;
            s_mn[j] = x * x + y * y;
        } else {
            s_pk[2 * j + 0] = 0.0f;
            s_pk[2 * j + 1] = 0.0f;
            s_mn[j] = 3.0e38f;
        }
    }
    __syncthreads();

    const int  lane  = tid & 31;
    const int  wave  = tid >> 5;
    const int  qbase = blockIdx.x * QPB + wave * 16;
    const bool lo    = lane < 16;

    // A fragment (16x4 f32), rows = queries.
    // K0 = -2x, K1 = -2y, K2 = 1.0 (picks up |m|^2 from B), K3 = 0.
    // Layout: VGPR0 = K0 (lanes 0-15) / K2 (lanes 16-31); VGPR1 = K1 / K3.
    int   q  = qbase + (lane & 15);
    int   qc = (q < N) ? q : (N - 1);
    float qx = ed[2 * qc + 0];
    float qy = ed[2 * qc + 1];
    bool  av = (q < N);
    v2f A;
    A.x = lo ? (av ? (-2.0f * qx) : 0.0f) : 1.0f;
    A.y = lo ? (av ? (-2.0f * qy) : 0.0f) : 0.0f;

    float bestv[8];
    int   besti[8];
#pragma unroll
    for (int i = 0; i < 8; ++i) { bestv[i] = 3.4e38f; besti[i] = 0; }

    int ntile = (M + 15) >> 4;
    if (ntile > (MPAD >> 4)) ntile = MPAD >> 4;

#pragma unroll 2
    for (int t = 0; t < ntile; ++t) {
        const int col = (t << 4) + (lane & 15);
        // One b64 load for (x,y), one b32 load for |m|^2.
        float2 p  = *(const float2*)&s_pk[2 * col];
        float  cn = s_mn[col];
        // B fragment (4x16 f32): K0 row = x, K1 row = y, K2 row = |m|^2, K3 = 0.
        v2f B;
        B.x = lo ? p.x : cn;
        B.y = lo ? p.y : 0.0f;
        // C = inline zero; D = |m|^2 - 2 q.m for the 16x16 tile.
        v8f C = {};
        v8f D = __builtin_amdgcn_wmma_f32_16x16x4_f32(
            /*neg_a=*/false, A, /*neg_b=*/false, B,
            /*c_mod=*/(short)0, C, /*reuse_a=*/false, /*reuse_b=*/false);
#pragma unroll
        for (int i = 0; i < 8; ++i) {
            float dv = D[i];
            if (dv < bestv[i]) { bestv[i] = dv; besti[i] = col; }
        }
    }

    // Reduce across the 16 columns of each row. D layout: VGPR i = row i for
    // lanes 0-15, row i+8 for lanes 16-31; xor offsets 8,4,2,1 stay in-half.
    // Tie-break on lower index (argmin first-occurrence semantics).
#pragma unroll
    for (int i = 0; i < 8; ++i) {
        float bv = bestv[i];
        int   bi = besti[i];
#pragma unroll
        for (int off = 8; off >= 1; off >>= 1) {
            float ov = __shfl_xor(bv, off, 32);
            int   oi = __shfl_xor(bi, off, 32);
            if (ov < bv || (ov == bv && oi < bi)) { bv = ov; bi = oi; }
        }
        bestv[i] = bv; besti[i] = bi;
    }

    // Lane 0 owns rows qbase+0..7, lane 16 owns rows qbase+8..15.
    if (lane == 0 || lane == 16) {
        const int rowbase = qbase + ((lane >> 4) << 3);
#pragma unroll
        for (int i = 0; i < 8; ++i) {
            int qq = rowbase + i;
            if (qq < N) {
                int j = besti[i];
                out[2 * qq + 0] = s_pk[2 * j + 0];
                out[2 * qq + 1] = s_pk[2 * j + 1];
                out[2 * N + qq] = ts[j];
            }
        }
    }
}

extern "C" void kernel_launch(void* const* d_in, const int* in_sizes, int n_in,
                              void* d_out, int out_size, void* d_ws, size_t ws_size,
                              hipStream_t stream) {
    const float* ed = (const float*)d_in[0];   // euclidean_data [N,2]
    const float* mp = (const float*)d_in[1];   // maze_points    [M,2]
    const float* ts = (const float*)d_in[2];   // ts_proj        [M]
    float* out = (float*)d_out;

    const int N = in_sizes[0] / 2;
    const int M = in_sizes[1] / 2;

    const int blocks = (N + QPB - 1) / QPB;
    LinearizationLayer_nn_wmma<<<blocks, 256, 0, stream>>>(ed, mp, ts, out, N, M);
}